// GCN_46102178955973
// MI455X (gfx1250) — compile-verified
//
#include <hip/hip_runtime.h>

typedef __attribute__((ext_vector_type(2))) float v2f;
typedef __attribute__((ext_vector_type(8))) float v8f;

// ---------------------------------------------------------------------------
// WMMA f32 16x16 tile GEMM: Y[M,N] = act(X[M,K]) @ W[K,N] (+ bias)
// One wave (32 lanes) computes one 16x16 tile of Y, stepping K by 4 with
// V_WMMA_F32_16X16X4_F32.  A layout: lane = M row (lane&15), lanes>=16 hold
// K+2/K+3 halves.  B/C/D: row striped across lanes within a VGPR.
// Requires K % 4 == 0 and N % 16 == 0 (true for all call sites here).
// ---------------------------------------------------------------------------
template<bool RELU_IN>
__global__ __launch_bounds__(128)
void wmma_mm_kernel(const float* __restrict__ X, const float* __restrict__ W,
                    const float* __restrict__ bias, float* __restrict__ Y,
                    int M, int K, int N) {
    const int lane   = threadIdx.x;                 // 0..31 (wave32)
    const int tile_m = blockIdx.x * blockDim.y + threadIdx.y;  // uniform per wave
    const int tile_n = blockIdx.y;
    const int mtiles = (M + 15) >> 4;
    if (tile_m >= mtiles) return;                   // whole-wave uniform exit

    const int mrow  = lane & 15;
    const int hi    = lane >> 4;                    // 0 or 1
    const int khalf = hi * 2;                       // K sub-offset {0,2}
    const int m     = tile_m * 16 + mrow;
    const int mc    = m < M ? m : (M - 1);          // clamp loads at M tail
    const int n0    = tile_n * 16;
    const int ncol  = mrow;

    const float* xrow = X + (size_t)mc * K + khalf;       // A: row mc, cols k+khalf..+1
    const float* wcol = W + (size_t)khalf * N + n0 + ncol; // B: row k+khalf, col n

    v8f acc = {};
    for (int k = 0; k < K; k += 4) {
        float a0 = xrow[k];
        float a1 = xrow[k + 1];
        if (RELU_IN) { a0 = fmaxf(a0, 0.f); a1 = fmaxf(a1, 0.f); }
        v2f a; a.x = a0; a.y = a1;
        v2f b;
        b.x = wcol[(size_t)k * N];
        b.y = wcol[(size_t)k * N + N];
        acc = __builtin_amdgcn_wmma_f32_16x16x4_f32(
            /*neg_a=*/false, a, /*neg_b=*/false, b,
            /*c_mod=*/(short)0, acc, /*reuse_a=*/false, /*reuse_b=*/false);
    }

    const float bv = bias ? bias[n0 + ncol] : 0.f;
    float* ycol = Y + n0 + ncol;
#pragma unroll
    for (int i = 0; i < 8; ++i) {
        int mm = tile_m * 16 + i + hi * 8;          // C/D layout: VGPR i -> rows i, i+8
        if (mm < M) ycol[(size_t)mm * N] = acc[i] + bv;
    }
}

// ---------------------------------------------------------------------------
// Edge scatter-add: out[dst[e]][c] += msg[src[e]][c]
// One thread per (edge, channel); consecutive lanes cover consecutive channels
// of the same edge -> coalesced gathers; atomics stay L2-resident (12.8 MB
// feature array << 192 MB L2).  C = 1<<LOGC.
// ---------------------------------------------------------------------------
template<int LOGC>
__global__ __launch_bounds__(256)
void scatter_add_kernel(const int* __restrict__ src, const int* __restrict__ dst,
                        const float* __restrict__ msg, float* __restrict__ out,
                        long long nwork) {
    const int CM = (1 << LOGC) - 1;
    long long t      = (long long)blockIdx.x * blockDim.x + threadIdx.x;
    long long stride = (long long)gridDim.x * blockDim.x;
    for (; t < nwork; t += stride) {
        long long e = t >> LOGC;
        int c = (int)(t & CM);
        int s = src[e];
        int d = dst[e];
        float v = msg[((size_t)s << LOGC) + c];
        atomicAdd(out + (((size_t)d) << LOGC) + c, v);
    }
}

// ---------------------------------------------------------------------------
// Graph pooling with fused ReLU: pool[batch[n]][c] += max(h[n][c], 0), C = 32
// ---------------------------------------------------------------------------
__global__ __launch_bounds__(256)
void pool_relu_kernel(const float* __restrict__ h, const int* __restrict__ batch,
                      float* __restrict__ pool, long long nwork) {
    long long t      = (long long)blockIdx.x * blockDim.x + threadIdx.x;
    long long stride = (long long)gridDim.x * blockDim.x;
    for (; t < nwork; t += stride) {
        long long n = t >> 5;
        int c = (int)(t & 31);
        float v = fmaxf(h[t], 0.f);
        atomicAdd(pool + (((size_t)batch[n]) << 5) + c, v);
    }
}

__global__ __launch_bounds__(256)
void zero_kernel(float* __restrict__ p, int n) {
    int t = blockIdx.x * blockDim.x + threadIdx.x;
    if (t < n) p[t] = 0.f;
}

extern "C" void kernel_launch(void* const* d_in, const int* in_sizes, int n_in,
                              void* d_out, int out_size, void* d_ws, size_t ws_size,
                              hipStream_t stream) {
    const float* x     = (const float*)d_in[0];
    const int*   ei    = (const int*)d_in[1];   // [2, E] int32
    const int*   batch = (const int*)d_in[2];
    const float* W1r = (const float*)d_in[3];
    const float* W1n = (const float*)d_in[4];
    const float* b1  = (const float*)d_in[5];
    const float* W2r = (const float*)d_in[6];
    const float* W2n = (const float*)d_in[7];
    const float* b2  = (const float*)d_in[8];
    const float* W3r = (const float*)d_in[9];
    const float* W3n = (const float*)d_in[10];
    const float* b3  = (const float*)d_in[11];
    const float* Wlin = (const float*)d_in[12];
    const float* blin = (const float*)d_in[13];

    const int       Nn = in_sizes[0] / 128;     // 100000 nodes
    const long long E  = in_sizes[1] / 2;       // 3.2M edges
    const int       G  = out_size / 64;         // 1000 graphs

    const int* src = ei;
    const int* dst = ei + E;

    // Workspace: three 100k x 32 f32 node buffers + pool (fits in d_ws).
    float* wsA  = (float*)d_ws;                 // nbr-transformed messages
    float* wsB  = wsA + (size_t)Nn * 32;        // node features (ping)
    float* wsC  = wsB + (size_t)Nn * 32;        // node features (pong)
    float* pool = wsC + (size_t)Nn * 32;        // [G, 32]

    const dim3 blk(32, 4);                      // 4 waves/block, one 16x16 tile each
    auto mt = [](int M) { return (M + 15) / 16; };
    auto gx = [&](int M) { return (unsigned)((mt(M) + 3) / 4); };

    const int SB = 256;
    auto sgrid = [&](long long nw) {
        long long b = (nw + SB - 1) / SB;
        return (unsigned)(b > (1ll << 22) ? (1ll << 22) : b);
    };

    // ---- Layer 1: 128 -> 16 ----------------------------------------------
    {
        dim3 g(gx(Nn), 1);  // N = 16 -> one tile column
        wmma_mm_kernel<false><<<g, blk, 0, stream>>>(x, W1n, nullptr, wsA, Nn, 128, 16);
        wmma_mm_kernel<false><<<g, blk, 0, stream>>>(x, W1r, b1,      wsB, Nn, 128, 16);
        long long nw = E * 16;
        scatter_add_kernel<4><<<sgrid(nw), SB, 0, stream>>>(src, dst, wsA, wsB, nw);
    }
    // ---- Layer 2: 16 -> 32 (ReLU folded into A load) ---------------------
    {
        dim3 g(gx(Nn), 2);
        wmma_mm_kernel<true><<<g, blk, 0, stream>>>(wsB, W2n, nullptr, wsA, Nn, 16, 32);
        wmma_mm_kernel<true><<<g, blk, 0, stream>>>(wsB, W2r, b2,      wsC, Nn, 16, 32);
        long long nw = E * 32;
        scatter_add_kernel<5><<<sgrid(nw), SB, 0, stream>>>(src, dst, wsA, wsC, nw);
    }
    // ---- Layer 3: 32 -> 32 -----------------------------------------------
    {
        dim3 g(gx(Nn), 2);
        wmma_mm_kernel<true><<<g, blk, 0, stream>>>(wsC, W3n, nullptr, wsA, Nn, 32, 32);
        wmma_mm_kernel<true><<<g, blk, 0, stream>>>(wsC, W3r, b3,      wsB, Nn, 32, 32);
        long long nw = E * 32;
        scatter_add_kernel<5><<<sgrid(nw), SB, 0, stream>>>(src, dst, wsA, wsB, nw);
    }
    // ---- Global add-pool (with ReLU of layer-3 output) -------------------
    {
        int pn = G * 32;
        zero_kernel<<<(pn + SB - 1) / SB, SB, 0, stream>>>(pool, pn);
        long long nw = (long long)Nn * 32;
        pool_relu_kernel<<<sgrid(nw), SB, 0, stream>>>(wsB, batch, pool, nw);
    }
    // ---- Final linear: [G,32] @ [32,64] + blin -> d_out ------------------
    {
        dim3 g(gx(G), 4);   // N = 64 -> 4 tile columns
        wmma_mm_kernel<false><<<g, blk, 0, stream>>>(pool, Wlin, blin, (float*)d_out, G, 32, 64);
    }
}